// dissected_Conv2d_52321291599893
// MI455X (gfx1250) — compile-verified
//
#include <hip/hip_runtime.h>

typedef float v2f __attribute__((ext_vector_type(2)));
typedef float v8f __attribute__((ext_vector_type(8)));

#define IN_C   64
#define OUT_C  128
#define IMG_H  56
#define IMG_W  56
#define BATCH  4
#define TILE_H 8
#define TILE_W 8
#define IC_STEP 8
#define KTILE  72            // IC_STEP * 9 taps
#define KTOT   (IN_C * 9)    // 576

// xs tile: [IC_STEP][TILE_H+2][TILE_W+2] = 8 x 10 x 10
#define XS_IC   100
#define XS_ROW  10
#define XS_N    (IC_STEP * XS_IC)   // 800

// im2col constant offset for flat k index kk (ic_local*9 + kh*3 + kw)
__device__ __forceinline__ constexpr int coff(int kk) {
    return (kk / 9) * XS_IC + ((kk % 9) / 3) * XS_ROW + ((kk % 9) % 3);
}

__global__ __launch_bounds__(256)
void dissected_conv2d_wmma(const float* __restrict__ x,
                           const float* __restrict__ wgt,
                           const float* __restrict__ bias,
                           float* __restrict__ out) {
    __shared__ float ws[64][KTILE];   // 64 oc x 72 k      (18.0 KB)
    __shared__ float xs[XS_N];        // 8 ic x 10 x 10    ( 3.2 KB)

    const int tid    = threadIdx.x;
    const int lane   = tid & 31;
    const int wave   = tid >> 5;
    const int half   = lane >> 4;     // 0: K pair {0,1}; 1: K pair {2,3}
    const int lane16 = lane & 15;

    // ---- block tile position ----
    const int s      = blockIdx.x;        // 0..195 : batch * 7 * 7 spatial tiles
    const int bb     = s / 49;
    const int srem   = s % 49;
    const int h_base = (srem / 7) * TILE_H;
    const int w_base = (srem % 7) * TILE_W;
    const int oc_base = blockIdx.y * 64;

    // ---- wave tile within block: 4 oc-subtiles x 2 n-subtiles ----
    const int oc_sub = (wave >> 1) * 16;
    const int n_sub  = (wave & 1) * 32;

    // B-fragment spatial bases for the two 16-wide N tiles (N = lane16 per ISA layout)
    const int n0 = n_sub + lane16;
    const int n1 = n0 + 16;
    const int base0 = (n0 >> 3) * XS_ROW + (n0 & 7);
    const int base1 = (n1 >> 3) * XS_ROW + (n1 & 7);

    v8f acc0 = {};
    v8f acc1 = {};

    for (int ic0 = 0; ic0 < IN_C; ic0 += IC_STEP) {
        // ---- cooperative load: x tile with zero padding at image borders ----
        for (int idx = tid; idx < XS_N; idx += 256) {
            int ic_l = idx / XS_IC;
            int rem  = idx - ic_l * XS_IC;
            int hh   = rem / XS_ROW;
            int ww   = rem - hh * XS_ROW;
            int gh   = h_base + hh - 1;
            int gw   = w_base + ww - 1;
            float v = 0.0f;
            const float* src =
                &x[(((size_t)bb * IN_C + ic0 + ic_l) * IMG_H + gh) * IMG_W + gw];
            if ((unsigned)gh < IMG_H && (unsigned)gw < IMG_W) v = *src;
            xs[idx] = v;
            // speculative prefetch of the next ic-chunk (safe if OOB: dropped)
            __builtin_prefetch(src + IC_STEP * IMG_H * IMG_W, 0, 1);
        }
        // ---- cooperative load: weight tile ws[oc_l][k] ----
        for (int idx = tid; idx < 64 * KTILE; idx += 256) {
            int oc_l = idx / KTILE;
            int kk   = idx - oc_l * KTILE;
            ws[oc_l][kk] = wgt[(size_t)(oc_base + oc_l) * KTOT + ic0 * 9 + kk];
        }
        __syncthreads();

        const float* wrow = &ws[oc_sub + lane16][0];

        #pragma unroll
        for (int k0 = 0; k0 < KTILE; k0 += 4) {
            // A fragment: contiguous k pair from this lane's weight row (ds_load_b64)
            v2f a = *(const v2f*)(wrow + k0 + 2 * half);
            // B fragments: im2col gather; tap offsets fold to constants post-unroll,
            // one cndmask each for the lane-half K selection.
            const int oA = half ? coff(k0 + 2) : coff(k0 + 0);
            const int oB = half ? coff(k0 + 3) : coff(k0 + 1);
            v2f b0, b1;
            b0.x = xs[oA + base0];
            b0.y = xs[oB + base0];
            b1.x = xs[oA + base1];
            b1.y = xs[oB + base1];
            acc0 = __builtin_amdgcn_wmma_f32_16x16x4_f32(
                false, a, false, b0, (short)0, acc0, false, false);
            acc1 = __builtin_amdgcn_wmma_f32_16x16x4_f32(
                false, a, false, b1, (short)0, acc1, false, false);
        }
        __syncthreads();
    }

    // ---- epilogue: C/D layout VGPR r -> M = r + 8*half, N = lane16 ----
    const int h0 = h_base + (n0 >> 3), w0 = w_base + (n0 & 7);
    const int h1 = h_base + (n1 >> 3), w1 = w_base + (n1 & 7);
    #pragma unroll
    for (int r = 0; r < 8; ++r) {
        int oc = oc_base + oc_sub + r + 8 * half;
        float bv = bias[oc];
        out[(((size_t)bb * OUT_C + oc) * IMG_H + h0) * IMG_W + w0] = acc0[r] + bv;
        out[(((size_t)bb * OUT_C + oc) * IMG_H + h1) * IMG_W + w1] = acc1[r] + bv;
    }
}

extern "C" void kernel_launch(void* const* d_in, const int* in_sizes, int n_in,
                              void* d_out, int out_size, void* d_ws, size_t ws_size,
                              hipStream_t stream) {
    const float* x    = (const float*)d_in[0];
    const float* wgt  = (const float*)d_in[1];
    const float* bias = (const float*)d_in[2];
    float* out        = (float*)d_out;

    dim3 grid(BATCH * (IMG_H / TILE_H) * (IMG_W / TILE_W), OUT_C / 64);  // (196, 2)
    dim3 block(256);
    dissected_conv2d_wmma<<<grid, block, 0, stream>>>(x, wgt, bias, out);
}